// HybridAttentionMIL_new_111669149919
// MI455X (gfx1250) — compile-verified
//
#include <hip/hip_runtime.h>
#include <math.h>

#define NROWS 100000
#define LDIM  1024
#define GDIM  768
#define ATTH  8
#define NCOL  16                 // Wv(8) || Wu(8)
#define FUSED (LDIM + GDIM)      // 1792
#define TILES (NROWS / 16)       // 6250 (exact)
#define K3_BLOCKS 625
#define ROWS_PER_B3 (NROWS / K3_BLOCKS)  // 160
#define PMAX_BLOCKS 128

// workspace layout (float offsets)
#define WG_OFF   0                         // 1024*16 = 16384
#define S1_OFF   16384                     // 16 : colsum(g*W)
#define S2_OFF   16400                     // 16 : colsum(b*W)
#define SCAL_OFF 16416                     // [0]=gmax [1]=sum(exp)
#define PMAX_OFF 16448                     // 128 partial maxes
#define ARAW_OFF (PMAX_OFF + PMAX_BLOCKS)  // N
#define MEAN_OFF (ARAW_OFF + NROWS)        // N
#define RSTD_OFF (MEAN_OFF + NROWS)        // N
#define PT_OFF   (RSTD_OFF + NROWS)        // 625 partial T sums (pad 640)
#define PEMB_OFF (PT_OFF + 640)            // 625*1024 partial embeddings

typedef __attribute__((ext_vector_type(2))) float v2f;
typedef __attribute__((ext_vector_type(8))) float v8f;

// ---------------------------------------------------------------- prep: Wg = g*[Wv||Wu], column sums
__global__ __launch_bounds__(256)
void prep_kernel(const float* __restrict__ g, const float* __restrict__ b,
                 const float* __restrict__ Wv, const float* __restrict__ Wu,
                 float* __restrict__ ws) {
  int t = threadIdx.x;
  int c = t & 15;        // column 0..15
  int seg = t >> 4;      // 0..15, 64 k-values each
  float p1 = 0.f, p2 = 0.f;
  for (int k = seg * 64; k < seg * 64 + 64; ++k) {
    float w = (c < ATTH) ? Wv[k * 8 + c] : Wu[k * 8 + (c - ATTH)];
    float gw = g[k] * w;
    ws[WG_OFF + k * 16 + c] = gw;
    p1 += gw;
    p2 += b[k] * w;
  }
  __shared__ float l1[256], l2[256];
  l1[t] = p1; l2[t] = p2;
  __syncthreads();
  if (t < 16) {
    float s1 = 0.f, s2 = 0.f;
    for (int s = 0; s < 16; ++s) { s1 += l1[s * 16 + t]; s2 += l2[s * 16 + t]; }
    ws[S1_OFF + t] = s1;
    ws[S2_OFF + t] = s2;
  }
}

// ---------------------------------------------------------------- pass1: WMMA GEMM + LN stats + attention logit
// one wave = one 16-row tile; D = bag[16 rows] @ Wg[1024,16] via V_WMMA_F32_16X16X4_F32
__global__ __launch_bounds__(256)
void pass1_kernel(const float* __restrict__ bag, float* __restrict__ ws,
                  const float* __restrict__ bv, const float* __restrict__ bu,
                  const float* __restrict__ Wa, const float* __restrict__ ba) {
  __shared__ float ldsWg[LDIM * NCOL];  // 64 KB
  int tid = threadIdx.x;
  for (int i = tid; i < (LDIM * NCOL) / 4; i += 256)
    ((float4*)ldsWg)[i] = ((const float4*)(ws + WG_OFF))[i];
  __syncthreads();

  int wave = tid >> 5, lane = tid & 31;
  int tile = blockIdx.x * 8 + wave;
  if (tile >= TILES) return;            // wave-uniform: EXEC stays all-ones for WMMA
  int r0 = tile * 16;
  int mrow = lane & 15;                 // A: lane -> matrix row ; D/B: lane -> column
  int half = lane >> 4;                 // K-pair select
  int col  = mrow;

  const float* ap = bag + (size_t)(r0 + mrow) * LDIM + 2 * half;  // float2 per K-step
  const float* bp = ldsWg + (2 * half) * NCOL + col;

  v8f acc = {0.f, 0.f, 0.f, 0.f, 0.f, 0.f, 0.f, 0.f};
  float sum = 0.f, ssq = 0.f;
#pragma unroll 4
  for (int k0 = 0; k0 < LDIM; k0 += 4) {
    float2 a2 = *(const float2*)(ap + k0);
    v2f a; a.x = a2.x; a.y = a2.y;
    v2f b; b.x = bp[k0 * NCOL]; b.y = bp[k0 * NCOL + NCOL];
    sum += a2.x + a2.y;
    ssq += a2.x * a2.x + a2.y * a2.y;
    acc = __builtin_amdgcn_wmma_f32_16x16x4_f32(false, a, false, b, (short)0, acc,
                                                false, false);
  }

  // full row stats: combine lane L with lane L^16
  float sum_f = sum + __shfl_xor(sum, 16, 32);
  float ssq_f = ssq + __shfl_xor(ssq, 16, 32);
  float mean_me = sum_f * (1.0f / LDIM);
  float var_me  = ssq_f * (1.0f / LDIM) - mean_me * mean_me;
  float rstd_me = rsqrtf(var_me + 1e-5f);
  if (lane < 16) {
    ws[MEAN_OFF + r0 + lane] = mean_me;
    ws[RSTD_OFF + r0 + lane] = rstd_me;
  }

  float s1c   = ws[S1_OFF + col];
  float s2c   = ws[S2_OFF + col];
  float biasc = (col < ATTH) ? bv[col] : bu[col - ATTH];
  float wac   = Wa[col & 7];
  float ba0   = ba[0];

#pragma unroll
  for (int v = 0; v < 8; ++v) {
    int r = v + (half << 3);                      // D row for this vgpr/lane-half
    float mr = __shfl(mean_me, r, 32);
    float rr = __shfl(rstd_me, r, 32);
    // layernorm affine correction: rstd*(x@Wg) - rstd*m*colsum(gW) + colsum(bW) + bias
    float proj = rr * acc[v] - rr * mr * s1c + s2c + biasc;
    float act  = (col < ATTH) ? tanhf(proj) : 1.0f / (1.0f + expf(-proj));
    float gate = act * __shfl_xor(act, 8, 32);    // tanh*sigmoid pairing
    float contrib = gate * wac;
    contrib += __shfl_xor(contrib, 1, 32);
    contrib += __shfl_xor(contrib, 2, 32);
    contrib += __shfl_xor(contrib, 4, 32);        // sum over h=0..7 (valid in col 0 group)
    if (col == 0) ws[ARAW_OFF + r0 + r] = contrib + ba0;
  }
}

// ---------------------------------------------------------------- softmax reductions
__global__ __launch_bounds__(256)
void max_kernel(float* __restrict__ ws) {
  __shared__ float red[256];
  float m = -3.4e38f;
  for (int i = blockIdx.x * 256 + threadIdx.x; i < NROWS; i += gridDim.x * 256)
    m = fmaxf(m, ws[ARAW_OFF + i]);
  red[threadIdx.x] = m;
  __syncthreads();
  for (int s = 128; s > 0; s >>= 1) {
    if (threadIdx.x < s) red[threadIdx.x] = fmaxf(red[threadIdx.x], red[threadIdx.x + s]);
    __syncthreads();
  }
  if (threadIdx.x == 0) ws[PMAX_OFF + blockIdx.x] = red[0];
}

__global__ __launch_bounds__(256)
void sum_kernel(float* __restrict__ ws) {
  __shared__ float red[256];
  int t = threadIdx.x;
  float m = (t < PMAX_BLOCKS) ? ws[PMAX_OFF + t] : -3.4e38f;
  red[t] = m; __syncthreads();
  for (int s = 128; s > 0; s >>= 1) { if (t < s) red[t] = fmaxf(red[t], red[t + s]); __syncthreads(); }
  float gmax = red[0];
  __syncthreads();
  float s = 0.f;
  for (int i = t; i < NROWS; i += 256)
    s += expf((ws[ARAW_OFF + i] - gmax) * 10.0f);     // /TEMPERATURE
  red[t] = s; __syncthreads();
  for (int st = 128; st > 0; st >>= 1) { if (t < st) red[t] += red[t + st]; __syncthreads(); }
  if (t == 0) { ws[SCAL_OFF] = gmax; ws[SCAL_OFF + 1] = red[0]; }
}

// ---------------------------------------------------------------- pass2: alpha + weighted bag sum (deterministic partials)
__global__ __launch_bounds__(256)
void pass2_kernel(const float* __restrict__ bag, float* __restrict__ ws,
                  float* __restrict__ out_alpha) {
  int t = threadIdx.x;
  float gmax = ws[SCAL_OFF], S = ws[SCAL_OFF + 1];
  float invS = 1.0f / S;
  float invDen = 1.0f / (1.0f + (float)NROWS * 1e-10f);
  int i0 = blockIdx.x * ROWS_PER_B3;
  float a0 = 0.f, a1 = 0.f, a2 = 0.f, a3 = 0.f, tpart = 0.f;
  const float* xp = bag + (size_t)i0 * LDIM + t * 4;
  for (int r = 0; r < ROWS_PER_B3; ++r) {
    int i = i0 + r;
    float e = expf((ws[ARAW_OFF + i] - gmax) * 10.0f);
    float alpha = (e * invS + 1e-10f) * invDen;
    float rs = ws[RSTD_OFF + i];
    float w = alpha * rs;
    if (t == 0) { out_alpha[i] = alpha; tpart += w * ws[MEAN_OFF + i]; }
    float4 x = *(const float4*)(xp + (size_t)r * LDIM);
    a0 += w * x.x; a1 += w * x.y; a2 += w * x.z; a3 += w * x.w;
  }
  float4 o; o.x = a0; o.y = a1; o.z = a2; o.w = a3;
  *(float4*)(ws + PEMB_OFF + (size_t)blockIdx.x * LDIM + t * 4) = o;
  if (t == 0) ws[PT_OFF + blockIdx.x] = tpart;
}

// ---------------------------------------------------------------- finale: embedding, global LN, MLP, score
__global__ __launch_bounds__(256)
void final_kernel(const float* __restrict__ gf,
                  const float* __restrict__ lng, const float* __restrict__ lnb,
                  const float* __restrict__ lgg, const float* __restrict__ lgb,
                  const float* __restrict__ W1, const float* __restrict__ b1,
                  const float* __restrict__ W2, const float* __restrict__ b2,
                  float* __restrict__ ws, float* __restrict__ out_score) {
  __shared__ float fused[FUSED];
  __shared__ float red[256];
  __shared__ float hred[8][256];
  int t = threadIdx.x;

  // T = sum alpha*rstd*mean
  float tp = 0.f;
  for (int i = t; i < K3_BLOCKS; i += 256) tp += ws[PT_OFF + i];
  red[t] = tp; __syncthreads();
  for (int s = 128; s > 0; s >>= 1) { if (t < s) red[t] += red[t + s]; __syncthreads(); }
  float T = red[0];
  __syncthreads();

  // reduce partial embeddings: cols 4t..4t+3
  float c0 = 0.f, c1 = 0.f, c2 = 0.f, c3 = 0.f;
  for (int b = 0; b < K3_BLOCKS; ++b) {
    float4 p = *(const float4*)(ws + PEMB_OFF + (size_t)b * LDIM + t * 4);
    c0 += p.x; c1 += p.y; c2 += p.z; c3 += p.w;
  }
  int c = t * 4;
  fused[c]     = lng[c]     * (c0 - T) + lnb[c];
  fused[c + 1] = lng[c + 1] * (c1 - T) + lnb[c + 1];
  fused[c + 2] = lng[c + 2] * (c2 - T) + lnb[c + 2];
  fused[c + 3] = lng[c + 3] * (c3 - T) + lnb[c + 3];

  // global-feature layernorm (768 = 256*3)
  float gs = 0.f, gq = 0.f, gv0[3];
  for (int j = 0; j < 3; ++j) { float v = gf[t * 3 + j]; gv0[j] = v; gs += v; gq += v * v; }
  red[t] = gs; __syncthreads();
  for (int s = 128; s > 0; s >>= 1) { if (t < s) red[t] += red[t + s]; __syncthreads(); }
  float gm = red[0] * (1.0f / GDIM);
  __syncthreads();
  red[t] = gq; __syncthreads();
  for (int s = 128; s > 0; s >>= 1) { if (t < s) red[t] += red[t + s]; __syncthreads(); }
  float gvar = red[0] * (1.0f / GDIM) - gm * gm;
  float grs = rsqrtf(gvar + 1e-5f);
  __syncthreads();
  for (int j = 0; j < 3; ++j) {
    int idx = t * 3 + j;
    fused[LDIM + idx] = (gv0[j] - gm) * grs * lgg[idx] + lgb[idx];
  }
  __syncthreads();

  // MLP: h = leaky_relu(fused @ W1 + b1); score = h @ W2 + b2
  float h[8] = {0.f, 0.f, 0.f, 0.f, 0.f, 0.f, 0.f, 0.f};
  for (int k = t; k < FUSED; k += 256) {
    float xk = fused[k];
#pragma unroll
    for (int j = 0; j < 8; ++j) h[j] += xk * W1[k * 8 + j];
  }
  for (int j = 0; j < 8; ++j) hred[j][t] = h[j];
  __syncthreads();
  for (int s = 128; s > 0; s >>= 1) {
    if (t < s) { for (int j = 0; j < 8; ++j) hred[j][t] += hred[j][t + s]; }
    __syncthreads();
  }
  if (t == 0) {
    float sc = b2[0];
    for (int j = 0; j < 8; ++j) {
      float hj = hred[j][0] + b1[j];
      hj = (hj > 0.f) ? hj : 0.01f * hj;
      sc += hj * W2[j];
    }
    out_score[0] = sc;
  }
}

// ---------------------------------------------------------------- launch
extern "C" void kernel_launch(void* const* d_in, const int* in_sizes, int n_in,
                              void* d_out, int out_size, void* d_ws, size_t ws_size,
                              hipStream_t stream) {
  const float* bag = (const float*)d_in[0];
  const float* gf  = (const float*)d_in[1];
  const float* lng = (const float*)d_in[2];
  const float* lnb = (const float*)d_in[3];
  const float* lgg = (const float*)d_in[4];
  const float* lgb = (const float*)d_in[5];
  const float* Wv  = (const float*)d_in[6];
  const float* bv  = (const float*)d_in[7];
  const float* Wu  = (const float*)d_in[8];
  const float* bu  = (const float*)d_in[9];
  const float* Wa  = (const float*)d_in[10];
  const float* ba  = (const float*)d_in[11];
  const float* W1  = (const float*)d_in[12];
  const float* b1  = (const float*)d_in[13];
  const float* W2  = (const float*)d_in[14];
  const float* b2  = (const float*)d_in[15];
  float* out = (float*)d_out;   // [0]=score, [1..N]=alpha
  float* ws  = (float*)d_ws;

  prep_kernel<<<1, 256, 0, stream>>>(lng, lnb, Wv, Wu, ws);
  pass1_kernel<<<(TILES + 7) / 8, 256, 0, stream>>>(bag, ws, bv, bu, Wa, ba);
  max_kernel<<<PMAX_BLOCKS, 256, 0, stream>>>(ws);
  sum_kernel<<<1, 256, 0, stream>>>(ws);
  pass2_kernel<<<K3_BLOCKS, 256, 0, stream>>>(bag, ws, out + 1);
  final_kernel<<<1, 256, 0, stream>>>(gf, lng, lnb, lgg, lgb, W1, b1, W2, b2, ws, out);
}